// NodeNetwork_30425548325108
// MI455X (gfx1250) — compile-verified
//
#include <hip/hip_runtime.h>
#include <hip/hip_bf16.h>
#include <math.h>

#define B_   2
#define N_   4096
#define E_   16384
#define D_   64
#define OUT_ 64
#define K1_  192   // 3*D

typedef __attribute__((ext_vector_type(2))) float v2f;
typedef __attribute__((ext_vector_type(8))) float v8f;

// ---------------------------------------------------------------------------
// Kernel 1: recover sparse indices from dense one-hot incidence matrix.
// R is [B, N, E] row-major; exactly one nonzero (=1.0f) per (b, e) column.
// Streaming float4 pass: 536 MB -> this is the roofline-dominant cost.
// ---------------------------------------------------------------------------
__global__ void extract_idx_kernel(const float* __restrict__ R,
                                   int* __restrict__ idx) {
    const long total4 = (long)B_ * N_ * E_ / 4;
    const long stride = (long)gridDim.x * blockDim.x;
    for (long i4 = (long)blockIdx.x * blockDim.x + threadIdx.x; i4 < total4; i4 += stride) {
        const long flat = i4 * 4;
        const float4 v = ((const float4*)R)[i4];
        // N*E = 2^26, E = 2^14, N = 2^12 (all powers of two)
        const int b  = (int)(flat >> 26);
        const int n  = (int)((flat >> 14) & (N_ - 1));
        const int e0 = (int)(flat & (E_ - 1));
        int* o = idx + b * E_ + e0;
        if (v.x != 0.0f) o[0] = n;
        if (v.y != 0.0f) o[1] = n;
        if (v.z != 0.0f) o[2] = n;
        if (v.w != 0.0f) o[3] = n;
    }
}

// ---------------------------------------------------------------------------
// Kernel 2: edge-weighted gather/scatter.
//   mi[b, ri[e], :] += e_w[b,e] * X[b, ro[e], :]
//   mo[b, ro[e], :] += e_w[b,e] * X[b, ri[e], :]
// 64 threads per edge (one feature element each), f32 global atomics.
// ---------------------------------------------------------------------------
__global__ void scatter_edges_kernel(const float* __restrict__ X,
                                     const float* __restrict__ ew,
                                     const int* __restrict__ ri,
                                     const int* __restrict__ ro,
                                     float* __restrict__ mi,
                                     float* __restrict__ mo) {
    const int g = blockIdx.x * (blockDim.x >> 6) + (threadIdx.x >> 6);  // edge id
    const int d = threadIdx.x & 63;
    if (g >= B_ * E_) return;
    const int b   = g / E_;
    const float w = ew[g];
    const int rin = ri[g];
    const int ron = ro[g];
    const float* Xb = X + (long)b * N_ * D_;
    const float xo = Xb[(long)ron * D_ + d];
    const float xi = Xb[(long)rin * D_ + d];
    atomicAdd(mi + ((long)b * N_ + rin) * D_ + d, w * xo);
    atomicAdd(mo + ((long)b * N_ + ron) * D_ + d, w * xi);
}

// ---------------------------------------------------------------------------
// Kernel 3: fused MLP with V_WMMA_F32_16X16X4_F32 (fp32 in / fp32 acc).
// One wave (32 lanes) per 16-row tile of the 8192x192 feature matrix.
//   h   = tanh([mi|mo|X] @ W1 + b1)    (192 -> 64)
//   out = tanh(h @ W2 + b2)            (64 -> 64)
// A frag: lane m=L%16, VGPR j holds K = 2*(L/16)+j.
// B frag: lane n=L%16, VGPR j holds K = 2*(L/16)+j.
// C/D:    lane n=L%16, VGPR v holds M = v + 8*(L/16).
// ---------------------------------------------------------------------------
__global__ __launch_bounds__(32)
void mlp_wmma_kernel(const float* __restrict__ mi, const float* __restrict__ mo,
                     const float* __restrict__ X,
                     const float* __restrict__ W1, const float* __restrict__ b1,
                     const float* __restrict__ W2, const float* __restrict__ b2,
                     float* __restrict__ out) {
    __shared__ float As[16][196];   // 16 x 192 feature tile (stride padded: 196 % 64 == 4)
    __shared__ float Hs[16][68];    // 16 x 64 hidden tile

    const int lane = threadIdx.x;        // 0..31
    const int lo   = lane & 15;          // N / M index within fragment
    const int hi   = lane >> 4;          // lane-half selects K pair / M half
    const int row0 = blockIdx.x * 16;

    // Stage A = [mi | mo | X] for rows row0..row0+15 (coalesced float4 loads).
    for (int i = lane; i < 16 * 16; i += 32) {
        const int  r = i >> 4;
        const int  q = (i & 15) * 4;
        const long base = (long)(row0 + r) * D_ + q;
        *(float4*)&As[r][q]       = *(const float4*)(mi + base);
        *(float4*)&As[r][64 + q]  = *(const float4*)(mo + base);
        *(float4*)&As[r][128 + q] = *(const float4*)(X  + base);
    }
    __syncthreads();

    // ---- GEMM1: (16x192) @ (192x64), K in steps of 4 ----
    v8f acc[4] = {};
    for (int k0 = 0; k0 < K1_; k0 += 4) {
        v2f a;
        a.x = As[lo][k0 + 2 * hi + 0];
        a.y = As[lo][k0 + 2 * hi + 1];
#pragma unroll
        for (int c = 0; c < 4; ++c) {
            v2f bb;
            bb.x = W1[(k0 + 2 * hi + 0) * OUT_ + 16 * c + lo];
            bb.y = W1[(k0 + 2 * hi + 1) * OUT_ + 16 * c + lo];
            acc[c] = __builtin_amdgcn_wmma_f32_16x16x4_f32(
                false, a, false, bb, (short)0, acc[c], false, false);
        }
    }

    // bias + tanh, transpose through LDS for the next GEMM's A layout
#pragma unroll
    for (int c = 0; c < 4; ++c) {
        const float bias = b1[16 * c + lo];
#pragma unroll
        for (int v = 0; v < 8; ++v) {
            Hs[v + 8 * hi][16 * c + lo] = tanhf(acc[c][v] + bias);
        }
    }
    __syncthreads();

    // ---- GEMM2: (16x64) @ (64x64) ----
    v8f acc2[4] = {};
    for (int k0 = 0; k0 < OUT_; k0 += 4) {
        v2f a;
        a.x = Hs[lo][k0 + 2 * hi + 0];
        a.y = Hs[lo][k0 + 2 * hi + 1];
#pragma unroll
        for (int c = 0; c < 4; ++c) {
            v2f bb;
            bb.x = W2[(k0 + 2 * hi + 0) * OUT_ + 16 * c + lo];
            bb.y = W2[(k0 + 2 * hi + 1) * OUT_ + 16 * c + lo];
            acc2[c] = __builtin_amdgcn_wmma_f32_16x16x4_f32(
                false, a, false, bb, (short)0, acc2[c], false, false);
        }
    }

    // bias + tanh + store
#pragma unroll
    for (int c = 0; c < 4; ++c) {
        const float bias = b2[16 * c + lo];
#pragma unroll
        for (int v = 0; v < 8; ++v) {
            out[(long)(row0 + v + 8 * hi) * OUT_ + 16 * c + lo] =
                tanhf(acc2[c][v] + bias);
        }
    }
}

// ---------------------------------------------------------------------------
extern "C" void kernel_launch(void* const* d_in, const int* in_sizes, int n_in,
                              void* d_out, int out_size, void* d_ws, size_t ws_size,
                              hipStream_t stream) {
    const float* X  = (const float*)d_in[0];   // [B,N,D]
    const float* ew = (const float*)d_in[1];   // [B,E]
    const float* Ri = (const float*)d_in[2];   // [B,N,E]
    const float* Ro = (const float*)d_in[3];   // [B,N,E]
    const float* W1 = (const float*)d_in[4];   // [192,64]
    const float* b1 = (const float*)d_in[5];   // [64]
    const float* W2 = (const float*)d_in[6];   // [64,64]
    const float* b2 = (const float*)d_in[7];   // [64]
    float* out = (float*)d_out;                // [B,N,64]

    char* ws = (char*)d_ws;
    int*   ri_idx = (int*)ws;                                   // B*E ints
    int*   ro_idx = (int*)(ws + (size_t)B_ * E_ * sizeof(int)); // B*E ints
    float* mi = (float*)(ws + (size_t)2 * B_ * E_ * sizeof(int));
    float* mo = mi + (size_t)B_ * N_ * D_;

    // Zero the scatter accumulators (graph-capture-safe async memset).
    hipMemsetAsync(mi, 0, (size_t)2 * B_ * N_ * D_ * sizeof(float), stream);

    extract_idx_kernel<<<2048, 256, 0, stream>>>(Ri, ri_idx);
    extract_idx_kernel<<<2048, 256, 0, stream>>>(Ro, ro_idx);

    scatter_edges_kernel<<<(B_ * E_) / 4, 256, 0, stream>>>(X, ew, ri_idx, ro_idx, mi, mo);

    mlp_wmma_kernel<<<(B_ * N_) / 16, 32, 0, stream>>>(mi, mo, X, W1, b1, W2, b2, out);
}